// GConv_meta_27230092657370
// MI455X (gfx1250) — compile-verified
//
#include <hip/hip_runtime.h>

typedef float v2f __attribute__((ext_vector_type(2)));
typedef float v8f __attribute__((ext_vector_type(8)));

#define CIN 256
#define COUT 128
#define TILE_M 16
#define LDS_STRIDE (CIN + 4)   // +4 floats: row stride % 64 banks == 4 -> conflict-free

// ---------------------------------------------------------------------------
// K0: zero the output accumulator (scatter-add target)
// ---------------------------------------------------------------------------
__global__ __launch_bounds__(256) void zero_kernel(float4* __restrict__ out, int n4) {
    int i = blockIdx.x * blockDim.x + threadIdx.x;
    if (i < n4) out[i] = make_float4(0.f, 0.f, 0.f, 0.f);
}

// ---------------------------------------------------------------------------
// K1: fused dropout(mask1) + Linear via FP32 WMMA (V_WMMA_F32_16X16X4_F32).
// One block = 16 rows of x, all 128 output channels (8 waves x 16 couts).
// x tile staged through LDS once; each wave runs 64 k-step WMMAs.
// ---------------------------------------------------------------------------
__global__ __launch_bounds__(256) void gemm_wmma_kernel(
    const float* __restrict__ emb, const float* __restrict__ mask1,
    const float* __restrict__ W,   const float* __restrict__ b_fc,
    float* __restrict__ feat)
{
    __shared__ float xs[TILE_M * LDS_STRIDE];

    const int tid = threadIdx.x;
    const int m0  = blockIdx.x * TILE_M;

    // Cooperative masked load of x tile [16 x 256] into LDS (float4 granularity).
    const float4* emb4 = (const float4*)(emb   + (size_t)m0 * CIN);
    const float4* m14  = (const float4*)(mask1 + (size_t)m0 * CIN);
    #pragma unroll
    for (int j = 0; j < 4; ++j) {
        int idx4 = tid + j * 256;     // 0..1023 float4 chunks (16*256/4)
        int r    = idx4 >> 6;         // 64 float4 per row
        int c4   = idx4 & 63;
        float4 e = emb4[r * 64 + c4];
        float4 m = m14 [r * 64 + c4];
        float4 x = make_float4(e.x * m.x, e.y * m.y, e.z * m.z, e.w * m.w);
        *(float4*)&xs[r * LDS_STRIDE + c4 * 4] = x;
    }
    __syncthreads();

    const int wave = tid >> 5;        // 8 waves -> 8 cout tiles of 16
    const int lane = tid & 31;
    const int co0  = wave * 16;
    const int half = lane >> 4;       // 0: K offsets {0,1}; 1: K offsets {2,3}
    const int l16  = lane & 15;

    // C/D layout: VGPR i holds M = i + 8*half, N = l16. b_fc depends only on N.
    v8f c;
    {
        float cinit = b_fc[co0 + l16];
        #pragma unroll
        for (int i = 0; i < 8; ++i) c[i] = cinit;
    }

    // A fragment: lane -> row M=l16 of x tile, K pair selected by half.
    // B fragment: lane -> column N=l16 (cout co0+l16), W row-major in CIN so
    //             B[k][n] = W[co0+l16][k] -> contiguous float2 in k.
    const float* xrow = xs + l16 * LDS_STRIDE;
    const float* wrow = W + (size_t)(co0 + l16) * CIN;

    #pragma unroll 8
    for (int k = 0; k < CIN; k += 4) {
        int kk = k + half * 2;
        v2f a = *(const v2f*)&xrow[kk];   // ds_load_b64, bank-conflict-free
        v2f b = *(const v2f*)&wrow[kk];   // global_load_b64, L2-resident W
        c = __builtin_amdgcn_wmma_f32_16x16x4_f32(
                /*neg_a=*/false, a, /*neg_b=*/false, b,
                /*c_mod=*/(short)0, c, /*reuse_a=*/false, /*reuse_b=*/false);
    }

    // Store D tile: row m0 + i + 8*half, col co0 + l16.
    float* frow = feat + (size_t)(m0 + half * 8) * COUT + co0 + l16;
    #pragma unroll
    for (int i = 0; i < 8; ++i)
        frow[(size_t)i * COUT] = c[i];
}

// ---------------------------------------------------------------------------
// K2: COO gather-scale-scatter. 32 lanes per edge, float4 per lane,
// GLOBAL_ATOMIC_ADD_F32 (no return) into the accumulator.
// ---------------------------------------------------------------------------
__global__ __launch_bounds__(256) void scatter_kernel(
    const float* __restrict__ feat, const float* __restrict__ vals,
    const int* __restrict__ rows,   const int* __restrict__ cols,
    float* __restrict__ out, int E)
{
    int t = blockIdx.x * blockDim.x + threadIdx.x;
    int e = t >> 5;                 // 32 threads per edge (128 couts / 4)
    if (e >= E) return;
    int c4  = (t & 31) * 4;
    int src = cols[e];
    int dst = rows[e];
    float v = vals[e];
    float4 f = *(const float4*)(feat + (size_t)src * COUT + c4);
    float* o = out + (size_t)dst * COUT + c4;
    atomicAdd(o + 0, f.x * v);
    atomicAdd(o + 1, f.y * v);
    atomicAdd(o + 2, f.z * v);
    atomicAdd(o + 3, f.w * v);
}

// ---------------------------------------------------------------------------
// K3: epilogue  out = prelu((out + bias) * mask2)
// ---------------------------------------------------------------------------
__global__ __launch_bounds__(256) void epilogue_kernel(
    float* __restrict__ out, const float* __restrict__ bias,
    const float* __restrict__ mask2, const float* __restrict__ prelu_a, int n4)
{
    int i4 = blockIdx.x * blockDim.x + threadIdx.x;
    if (i4 >= n4) return;
    float a  = prelu_a[0];
    float4 o = ((float4*)out)[i4];
    float4 m = ((const float4*)mask2)[i4];
    int c    = (i4 * 4) & (COUT - 1);       // COUT power of two
    float4 b = *(const float4*)(bias + c);
    float t;
    t = (o.x + b.x) * m.x; o.x = t > 0.f ? t : a * t;
    t = (o.y + b.y) * m.y; o.y = t > 0.f ? t : a * t;
    t = (o.z + b.z) * m.z; o.z = t > 0.f ? t : a * t;
    t = (o.w + b.w) * m.w; o.w = t > 0.f ? t : a * t;
    ((float4*)out)[i4] = o;
}

// ---------------------------------------------------------------------------
// Launcher. Input order (setup_inputs dict): emb, vals, W, b_fc, bias,
// prelu_a, mask1, mask2, rows, cols. Output: [N, COUT] float32.
// ---------------------------------------------------------------------------
extern "C" void kernel_launch(void* const* d_in, const int* in_sizes, int n_in,
                              void* d_out, int out_size, void* d_ws, size_t ws_size,
                              hipStream_t stream) {
    const float* emb     = (const float*)d_in[0];
    const float* vals    = (const float*)d_in[1];
    const float* W       = (const float*)d_in[2];
    const float* b_fc    = (const float*)d_in[3];
    const float* bias    = (const float*)d_in[4];
    const float* prelu_a = (const float*)d_in[5];
    const float* mask1   = (const float*)d_in[6];
    const float* mask2   = (const float*)d_in[7];
    const int*   rows    = (const int*)d_in[8];
    const int*   cols    = (const int*)d_in[9];

    const int N = in_sizes[0] / CIN;    // 100000
    const int E = in_sizes[1];          // 1600000

    float* feat = (float*)d_ws;         // [N, COUT] scratch (51.2 MB)
    float* out  = (float*)d_out;        // [N, COUT]

    // K0: zero accumulator
    {
        int n4 = (N * COUT) / 4;
        zero_kernel<<<(n4 + 255) / 256, 256, 0, stream>>>((float4*)out, n4);
    }
    // K1: fused dropout + Linear (WMMA)
    {
        int grid = N / TILE_M;          // 6250 (N divisible by 16)
        gemm_wmma_kernel<<<grid, 256, 0, stream>>>(emb, mask1, W, b_fc, feat);
    }
    // K2: gather-scale-scatter
    {
        long long threads = (long long)E * 32;
        int grid = (int)((threads + 255) / 256);
        scatter_kernel<<<grid, 256, 0, stream>>>(feat, vals, rows, cols, out, E);
    }
    // K3: epilogue
    {
        int n4 = (N * COUT) / 4;
        epilogue_kernel<<<(n4 + 255) / 256, 256, 0, stream>>>(out, bias, mask2, prelu_a, n4);
    }
}